// SuperResolve_76304388980743
// MI455X (gfx1250) — compile-verified
//
#include <hip/hip_runtime.h>
#include <stdint.h>

// Problem dims (fixed by the reference setup_inputs)
#define A_FR 4
#define NB   4
#define CHN  3
#define H    256
#define W    256
#define SH   (2*H)
#define SW   (2*W)
#define HW   (H*W)

// Tiling: each thread computes one 2x2 output quad (the 4 hi-res pixels that
// share a low-res 3x3 neighborhood). 256 threads = 32x8 quads = 64x16 outputs.
#define QX   32
#define QY   8
#define LXT  (QX + 2)     // 34 low-res cols incl. halo
#define LYT  (QY + 2)     // 10 low-res rows incl. halo
#define LNT  (LXT*LYT)    // 340

// -0.5*log2(e) and 20*0.5*log2(e): exp(-0.5*clip(d,0,20)) == exp2(clip(-K*d, -20K, 0))
#define NEG_K    (-0.7213475204444817f)
#define NEG_K20  (-14.426950408889634f)

#if __has_builtin(__builtin_amdgcn_global_load_async_to_lds_b32) && __has_builtin(__builtin_amdgcn_s_wait_asynccnt)
#define USE_ASYNC_LDS 1
#else
#define USE_ASYNC_LDS 0
#endif

typedef __attribute__((address_space(1))) int glb_i32;
typedef __attribute__((address_space(3))) int lds_i32;

// Stage one dword global -> LDS: global_load_async_to_lds_b32 (ASYNCcnt-tracked,
// no VGPR round trip).
__device__ __forceinline__ void stage_b32(float* l, const float* g) {
#if USE_ASYNC_LDS
  __builtin_amdgcn_global_load_async_to_lds_b32(
      (glb_i32*)(uintptr_t)g,              // generic->as1: numerically identical
      (lds_i32*)(uint32_t)(uintptr_t)l,    // LDS offset is the low 32 bits of the address
      0, 0);
#else
  *l = *g;
#endif
}

__device__ __forceinline__ float fast_exp2(float x) {
#if __has_builtin(__builtin_amdgcn_exp2f)
  return __builtin_amdgcn_exp2f(x);
#else
  return exp2f(x);
#endif
}

__global__ __launch_bounds__(256)
void superres_rbf_kernel(const float* __restrict__ alts,     // (A,n,c,h,w)
                         const float* __restrict__ offsets,  // (A,n,2,h,w)
                         const float* __restrict__ qs,       // (A,n,h,w)
                         const float* __restrict__ o11,      // (n,sh,sw)
                         const float* __restrict__ o12,
                         const float* __restrict__ o21,
                         const float* __restrict__ o22,
                         float* __restrict__ out)            // (n,c,sh,sw)
{
  // LDS tile per (frame, low-res px): [off_r, off_c, q, alt_c0] + [alt_c1, alt_c2]
  __shared__ float4 tA[A_FR][LNT];   // 21760 B
  __shared__ float2 tB[A_FR][LNT];   // 10880 B

  const int tid = threadIdx.x;
  const int n   = blockIdx.z;
  const int ty0 = blockIdx.y * QY;   // low-res tile origin (quad == low-res px)
  const int tx0 = blockIdx.x * QX;
  const int ry0 = ty0 - 1;           // incl. halo
  const int rx0 = tx0 - 1;

  // ---- Stage the low-res halo tile for all 4 frames (clamped coords; OOB
  // neighbors are masked to zero weight later, matching the reference) ----
  for (int t = tid; t < A_FR * LNT; t += 256) {
    const int a  = t / LNT;
    const int i  = t - a * LNT;
    const int lr = i / LXT;
    const int lc = i - lr * LXT;
    const int r  = min(max(ry0 + lr, 0), H - 1);
    const int c  = min(max(rx0 + lc, 0), W - 1);
    const int an = a * NB + n;
    const float* offp = offsets + (size_t)(an * 2) * HW   + (size_t)r * W + c;
    const float* qp   = qs      + (size_t)an * HW         + (size_t)r * W + c;
    const float* ap   = alts    + (size_t)(an * CHN) * HW + (size_t)r * W + c;
    float* da = (float*)&tA[a][i];
    float* db = (float*)&tB[a][i];
    stage_b32(da + 0, offp);          // off_r
    stage_b32(da + 1, offp + HW);     // off_c
    stage_b32(da + 2, qp);            // q
    stage_b32(da + 3, ap);            // alt c0
    stage_b32(db + 0, ap + HW);       // alt c1
    stage_b32(db + 1, ap + 2 * HW);   // alt c2
  }
#if USE_ASYNC_LDS
  __builtin_amdgcn_s_wait_asynccnt(0);
#endif
  __syncthreads();

  // ---- Per-quad compute: one thread per 2x2 output block, all 3 channels ----
  const int qx = tid & (QX - 1);     // 0..31
  const int qy = tid >> 5;           // 0..7
  const int y0 = ty0 + qy;           // low-res center of this quad
  const int x0 = tx0 + qx;
  const int lbase = (qy + 1) * LXT + (qx + 1);

  // Edge validity, hoisted: neighbors go OOB only at the 4 grid edges.
  const bool yok[3] = { y0 > 0, true, y0 < H - 1 };
  const bool xok[3] = { x0 > 0, true, x0 < W - 1 };

  // Covariance coefficients for the 4 output pixels (rows p=0,1; cols q=0,1),
  // pre-scaled by -0.5*log2(e) so the exp argument needs no extra multiply.
  const size_t base0 = ((size_t)n * SH + 2 * y0) * SW + 2 * x0;
  const size_t base1 = base0 + SW;
  const float2 c11r0 = *(const float2*)(o11 + base0);
  const float2 c11r1 = *(const float2*)(o11 + base1);
  const float2 t12r0 = *(const float2*)(o12 + base0);
  const float2 t12r1 = *(const float2*)(o12 + base1);
  const float2 t21r0 = *(const float2*)(o21 + base0);
  const float2 t21r1 = *(const float2*)(o21 + base1);
  const float2 c22r0 = *(const float2*)(o22 + base0);
  const float2 c22r1 = *(const float2*)(o22 + base1);
  const float n11r0x = NEG_K * c11r0.x,           n11r0y = NEG_K * c11r0.y;
  const float n11r1x = NEG_K * c11r1.x,           n11r1y = NEG_K * c11r1.y;
  const float n12r0x = NEG_K * (t12r0.x + t21r0.x), n12r0y = NEG_K * (t12r0.y + t21r0.y);
  const float n12r1x = NEG_K * (t12r1.x + t21r1.x), n12r1y = NEG_K * (t12r1.y + t21r1.y);
  const float n22r0x = NEG_K * c22r0.x,           n22r0y = NEG_K * c22r0.y;
  const float n22r1x = NEG_K * c22r1.x,           n22r1y = NEG_K * c22r1.y;

  float ws[2][2]  = {{0.f, 0.f}, {0.f, 0.f}};
  float ac[3][2][2] = {};

#pragma unroll 2
  for (int a = 0; a < A_FR; ++a) {
#pragma unroll
    for (int k = 0; k < 9; ++k) {
      const int di = k / 3 - 1;
      const int dj = k - (k / 3) * 3 - 1;
      const bool inb = yok[di + 1] && xok[dj + 1];   // folds per unrolled k
      const int li = lbase + di * LXT + dj;
      const float4 v0 = tA[a][li];    // off_r, off_c, q, alt0
      const float2 v1 = tB[a][li];    // alt1, alt2

      // Displacements in low-res units (shared across the quad):
      //   dy_p = di + off_r - 0.5p ; dx_q = dj + off_c - 0.5q ; valid iff |d|<=2
      const float dy0 = (float)di + v0.x;
      const float dy1 = dy0 - 0.5f;
      const float dx0 = (float)dj + v0.y;
      const float dx1 = dx0 - 0.5f;
      // Row masks pre-ANDed with edge validity; column masks standalone.
      const bool okr0 = inb && (__builtin_fabsf(dy0) <= 2.0f);
      const bool okr1 = inb && (__builtin_fabsf(dy1) <= 2.0f);
      const bool ox0b = __builtin_fabsf(dx0) <= 2.0f;
      const bool ox1b = __builtin_fabsf(dx1) <= 2.0f;

      // Shared products
      const float dx0s = dx0 * dx0, dx1s = dx1 * dx1;
      const float dy0s = dy0 * dy0, dy1s = dy1 * dy1;
      const float p00 = dx0 * dy0, p10 = dx1 * dy0;
      const float p01 = dx0 * dy1, p11 = dx1 * dy1;

      auto wgt = [&](float dxs, float dys, float pcr,
                     float a11, float a12, float a22, bool ok) -> float {
        // dist_neg = -K * rbf_dist ; clip(d,0,20) * -K == clamp(dist_neg, -20K, 0)
        float dn = __builtin_fmaf(pcr, a12, dxs * a11);
        dn = __builtin_fmaf(dys, a22, dn);
        dn = fminf(fmaxf(dn, NEG_K20), 0.0f);
        const float rbf = fast_exp2(dn);   // single v_exp_f32
        return ok ? rbf * v0.z : 0.0f;
      };

      const float w00 = wgt(dx0s, dy0s, p00, n11r0x, n12r0x, n22r0x, okr0 && ox0b);
      const float w01 = wgt(dx1s, dy0s, p10, n11r0y, n12r0y, n22r0y, okr0 && ox1b);
      const float w10 = wgt(dx0s, dy1s, p01, n11r1x, n12r1x, n22r1x, okr1 && ox0b);
      const float w11 = wgt(dx1s, dy1s, p11, n11r1y, n12r1y, n22r1y, okr1 && ox1b);

      ws[0][0] += w00; ws[0][1] += w01; ws[1][0] += w10; ws[1][1] += w11;
      ac[0][0][0] = __builtin_fmaf(v0.w, w00, ac[0][0][0]);
      ac[0][0][1] = __builtin_fmaf(v0.w, w01, ac[0][0][1]);
      ac[0][1][0] = __builtin_fmaf(v0.w, w10, ac[0][1][0]);
      ac[0][1][1] = __builtin_fmaf(v0.w, w11, ac[0][1][1]);
      ac[1][0][0] = __builtin_fmaf(v1.x, w00, ac[1][0][0]);
      ac[1][0][1] = __builtin_fmaf(v1.x, w01, ac[1][0][1]);
      ac[1][1][0] = __builtin_fmaf(v1.x, w10, ac[1][1][0]);
      ac[1][1][1] = __builtin_fmaf(v1.x, w11, ac[1][1][1]);
      ac[2][0][0] = __builtin_fmaf(v1.y, w00, ac[2][0][0]);
      ac[2][0][1] = __builtin_fmaf(v1.y, w01, ac[2][0][1]);
      ac[2][1][0] = __builtin_fmaf(v1.y, w10, ac[2][1][0]);
      ac[2][1][1] = __builtin_fmaf(v1.y, w11, ac[2][1][1]);
    }
  }

  // Reciprocal (+1 Newton step) per pixel instead of 3 IEEE divides each.
  float inv[2][2];
#pragma unroll
  for (int p = 0; p < 2; ++p)
#pragma unroll
    for (int q = 0; q < 2; ++q) {
      const float wv = ws[p][q];
#if __has_builtin(__builtin_amdgcn_rcpf)
      float r = __builtin_amdgcn_rcpf(wv);
      r = r * (2.0f - wv * r);
#else
      const float r = 1.0f / wv;
#endif
      inv[p][q] = r;
    }

  // Coalesced float2 stores: the two columns of a quad are contiguous.
#pragma unroll
  for (int ch = 0; ch < CHN; ++ch) {
    float* row0 = out + ((size_t)(n * CHN + ch) * SH + 2 * y0) * SW + 2 * x0;
    *(float2*)(row0)      = make_float2(ac[ch][0][0] * inv[0][0],
                                        ac[ch][0][1] * inv[0][1]);
    *(float2*)(row0 + SW) = make_float2(ac[ch][1][0] * inv[1][0],
                                        ac[ch][1][1] * inv[1][1]);
  }
}

extern "C" void kernel_launch(void* const* d_in, const int* in_sizes, int n_in,
                              void* d_out, int out_size, void* d_ws, size_t ws_size,
                              hipStream_t stream) {
  const float* alts    = (const float*)d_in[0];
  const float* offsets = (const float*)d_in[1];
  const float* qs      = (const float*)d_in[2];
  const float* o11     = (const float*)d_in[3];
  const float* o12     = (const float*)d_in[4];
  const float* o21     = (const float*)d_in[5];
  const float* o22     = (const float*)d_in[6];
  float* out = (float*)d_out;

  dim3 grid(W / QX, H / QY, NB);     // (8, 32, 4) blocks, 256 thr = 8 waves each
  dim3 block(256);
  hipLaunchKernelGGL(superres_rbf_kernel, grid, block, 0, stream,
                     alts, offsets, qs, o11, o12, o21, o22, out);
}